// Linear_SpatialAttention_11897059410211
// MI455X (gfx1250) — compile-verified
//
#include <hip/hip_runtime.h>

// Fused: qkv 1x1 conv -> per-head linear attention (softmax-d on q, softmax-n on k,
// k@v^T then ctx^T@q) -> out 1x1 conv + bias -> channel LayerNorm.
// One workgroup (256 thr = 8 wave32) per (batch, 8x8 spatial patch). All GEMMs use
// V_WMMA_F32_16X16X4_F32; operand fragments are batch-loaded and shared across
// output tiles for ILP (multi-accumulator k-loops).

typedef float v2f __attribute__((ext_vector_type(2)));
typedef float v8f __attribute__((ext_vector_type(8)));

__device__ __forceinline__ v8f wmma4(v2f a, v2f b, v8f c) {
  // D = A(16x4 f32) * B(4x16 f32) + C(16x16 f32)
  return __builtin_amdgcn_wmma_f32_16x16x4_f32(false, a, false, b, (short)0, c, false, false);
}

// LDS layout (floats). XT aliased as AOT, KB aliased as YT (lifetimes disjoint).
#define OFF_XT   0       // [64][132]  x^T (pixel-major)   / later AOT attn-out^T
#define OFF_QT   8448    // [4][64][34] q^T per head
#define OFF_KB   17152   // [128][68]  k rows              / later YT y^T [64][132]
#define OFF_VB   25856   // [128][68]  v rows
#define OFF_CT   34560   // [4][32][34] context^T
#define OFF_BO   38912   // [128] bout
#define OFF_GS   39040   // [128] g
#define OFF_MEAN 39168   // [64]
#define OFF_RSTD 39232   // [64]
#define SMEM_FLOATS 39296
#define SMEM_BYTES  (SMEM_FLOATS * 4)

__global__ __launch_bounds__(256, 1)
void fused_linattn_kernel(const float* __restrict__ x,
                          const float* __restrict__ Wqkv,
                          const float* __restrict__ Wout,
                          const float* __restrict__ bout,
                          const float* __restrict__ g,
                          float* __restrict__ out)
{
  extern __shared__ float sm[];
  float* XT    = sm + OFF_XT;
  float* QT    = sm + OFF_QT;
  float* KB    = sm + OFF_KB;
  float* VB    = sm + OFF_VB;
  float* CT    = sm + OFF_CT;
  float* BO    = sm + OFF_BO;
  float* GS    = sm + OFF_GS;
  float* MEANv = sm + OFF_MEAN;
  float* RSTDv = sm + OFF_RSTD;
  float* AOT = XT;   // [64][132] attn-out^T (pixel-major, 128 channels)
  float* YT  = KB;   // [64][132] y^T

  const int tid  = threadIdx.x;
  const int wave = tid >> 5;
  const int lane = tid & 31;
  const int l16  = lane & 15;
  const int hi   = lane >> 4;     // half-wave: K pairs {0,1} vs {2,3}
  const int kofs = 2 * hi;

  const int b  = blockIdx.y;
  const int ph = blockIdx.x >> 3;
  const int pw = blockIdx.x & 7;
  const long gbase = (long)b * 524288 + (long)(ph * 8) * 64 + (long)(pw * 8);

  // ---- Load x patch: x[b, c, ph*8+r, pw*8 .. +7] -> XT[n=r*8+col][c]
  for (int it = 0; it < 4; ++it) {
    int p = tid + 256 * it;          // 1024 (c, r) pairs
    int c = p >> 3, r = p & 7;
    const float4* src = (const float4*)(x + gbase + (long)c * 4096 + (long)r * 64);
    float4 a0 = src[0], a1 = src[1];
    float vals[8] = {a0.x, a0.y, a0.z, a0.w, a1.x, a1.y, a1.z, a1.w};
#pragma unroll
    for (int col = 0; col < 8; ++col)
      XT[(r * 8 + col) * 132 + c] = vals[col];
  }
  if (tid < 128) { BO[tid] = bout[tid]; GS[tid] = g[tid]; }
  __syncthreads();

  // ---- Stage 1: qkv[384x64] = Wqkv[384x128] @ X[128x64]
  // Single k-loop, 12 accumulators: 3 row-blocks (q/k/v: mt = wave + 8j) x 4 N-tiles.
  // B fragments (XT) are shared across the 3 row-blocks.
  {
    v8f acc[12] = {};   // acc[j*4 + nt]
    const float* Wbase = Wqkv + (wave * 16 + l16) * 128 + kofs; // j stride = 16384 floats
    const float* Xbase = &XT[l16 * 132 + kofs];                 // nt stride = 2112 floats
    for (int k = 0; k < 128; k += 4) {
      v2f a0 = *(const v2f*)(Wbase + k);
      v2f a1 = *(const v2f*)(Wbase + 16384 + k);
      v2f a2 = *(const v2f*)(Wbase + 32768 + k);
      v2f b0 = *(const v2f*)(Xbase + 0 * 2112 + k);
      v2f b1 = *(const v2f*)(Xbase + 1 * 2112 + k);
      v2f b2 = *(const v2f*)(Xbase + 2 * 2112 + k);
      v2f b3 = *(const v2f*)(Xbase + 3 * 2112 + k);
      acc[0]  = wmma4(a0, b0, acc[0]);  acc[1]  = wmma4(a0, b1, acc[1]);
      acc[2]  = wmma4(a0, b2, acc[2]);  acc[3]  = wmma4(a0, b3, acc[3]);
      acc[4]  = wmma4(a1, b0, acc[4]);  acc[5]  = wmma4(a1, b1, acc[5]);
      acc[6]  = wmma4(a1, b2, acc[6]);  acc[7]  = wmma4(a1, b3, acc[7]);
      acc[8]  = wmma4(a2, b0, acc[8]);  acc[9]  = wmma4(a2, b1, acc[9]);
      acc[10] = wmma4(a2, b2, acc[10]); acc[11] = wmma4(a2, b3, acc[11]);
    }
#pragma unroll
    for (int j = 0; j < 3; ++j) {
      const int mt = wave + 8 * j;
#pragma unroll
      for (int nt = 0; nt < 4; ++nt) {
        int nn = nt * 16 + l16;
#pragma unroll
        for (int i = 0; i < 8; ++i) {
          int mm = mt * 16 + i + 8 * hi;           // global output row (o)
          float val = acc[j * 4 + nt][i];
          if (j == 0) {        // q -> QT[h][n][d]
            QT[(mm >> 5) * 2176 + nn * 34 + (mm & 31)] = val;
          } else if (j == 1) { // k -> KB[c][n]
            KB[(mm - 128) * 68 + nn] = val;
          } else {             // v -> VB[c][n]
            VB[(mm - 256) * 68 + nn] = val;
          }
        }
      }
    }
  }
  __syncthreads();

  // ---- softmax over d for q (per (head, pixel) column), fold in scale = 32^-0.5
  {
    int h = tid >> 6, n = tid & 63;
    float* row = &QT[h * 2176 + n * 34];
    float mx = row[0];
#pragma unroll
    for (int d = 1; d < 32; ++d) mx = fmaxf(mx, row[d]);
    float e[32]; float s = 0.f;
#pragma unroll
    for (int d = 0; d < 32; ++d) { e[d] = expf(row[d] - mx); s += e[d]; }
    float inv = 0.17677669529663687f / s;
#pragma unroll
    for (int d = 0; d < 32; ++d) row[d] = e[d] * inv;
  }
  // ---- softmax over n for k (per (head, channel) row)
  if (tid < 128) {
    float* row = &KB[tid * 68];
    float mx = row[0];
    for (int n2 = 1; n2 < 64; ++n2) mx = fmaxf(mx, row[n2]);
    float s = 0.f;
    for (int n2 = 0; n2 < 64; ++n2) { float t = expf(row[n2] - mx); row[n2] = t; s += t; }
    float inv = 1.f / s;
    for (int n2 = 0; n2 < 64; ++n2) row[n2] *= inv;
  }
  __syncthreads();

  // ---- ctx[d,e] = sum_n k[d,n] * v[e,n]  (per head, 32x32, K=64) -> store CT = ctx^T
  // B fragment shared across the two M-tiles.
  {
    int h = wave >> 1, sub = wave & 1;
    v8f acc0 = {}, acc1 = {};
    const float* A0 = &KB[(h * 32 + l16) * 68 + kofs];
    const float* A1 = A0 + 16 * 68;
    const float* Bp = &VB[(h * 32 + sub * 16 + l16) * 68 + kofs];
#pragma unroll
    for (int k = 0; k < 64; k += 4) {
      v2f b  = *(const v2f*)(Bp + k);
      v2f a0 = *(const v2f*)(A0 + k);
      v2f a1 = *(const v2f*)(A1 + k);
      acc0 = wmma4(a0, b, acc0);
      acc1 = wmma4(a1, b, acc1);
    }
#pragma unroll
    for (int i = 0; i < 8; ++i) {
      int e2 = sub * 16 + l16;
      CT[h * 1088 + e2 * 34 + ( 0 + i + 8 * hi)] = acc0[i];
      CT[h * 1088 + e2 * 34 + (16 + i + 8 * hi)] = acc1[i];
    }
  }
  __syncthreads();

  // ---- out[e,n] = sum_d ctx[d,e] * q[d,n]  (per head, 32x64, K=32) -> AOT[n][h*32+e]
  // 2 A-loads + 2 B-loads -> 4 WMMAs per k-step.
  {
    int h = wave >> 1, sub = wave & 1;
    v8f acc[4] = {};   // acc[mt*2 + q2], nt = sub*2 + q2
    const float* A0 = &CT[h * 1088 + l16 * 34 + kofs];
    const float* A1 = A0 + 16 * 34;
    const float* B0 = &QT[h * 2176 + (sub * 32 + l16) * 34 + kofs];
    const float* B1 = B0 + 16 * 34;
#pragma unroll
    for (int k = 0; k < 32; k += 4) {
      v2f a0 = *(const v2f*)(A0 + k);
      v2f a1 = *(const v2f*)(A1 + k);
      v2f b0 = *(const v2f*)(B0 + k);
      v2f b1 = *(const v2f*)(B1 + k);
      acc[0] = wmma4(a0, b0, acc[0]);
      acc[1] = wmma4(a0, b1, acc[1]);
      acc[2] = wmma4(a1, b0, acc[2]);
      acc[3] = wmma4(a1, b1, acc[3]);
    }
#pragma unroll
    for (int mt = 0; mt < 2; ++mt) {
#pragma unroll
      for (int q2 = 0; q2 < 2; ++q2) {
        int nt = sub * 2 + q2;
        int nn = nt * 16 + l16;
#pragma unroll
        for (int i = 0; i < 8; ++i) {
          int e2 = mt * 16 + i + 8 * hi;
          AOT[nn * 132 + h * 32 + e2] = acc[mt * 2 + q2][i];
        }
      }
    }
  }
  __syncthreads();

  // ---- y[128x64] = Wout[128x128] @ AO[128x64] + bout  -> YT[n][o]
  {
    const float* Wrow  = Wout + (wave * 16 + l16) * 128 + kofs;
    const float* Abase = &AOT[l16 * 132 + kofs];   // nt stride = 2112 floats
    v8f acc[4] = {};
    for (int k = 0; k < 128; k += 4) {
      v2f a  = *(const v2f*)(Wrow + k);
      v2f b0 = *(const v2f*)(Abase + 0 * 2112 + k);
      v2f b1 = *(const v2f*)(Abase + 1 * 2112 + k);
      v2f b2 = *(const v2f*)(Abase + 2 * 2112 + k);
      v2f b3 = *(const v2f*)(Abase + 3 * 2112 + k);
      acc[0] = wmma4(a, b0, acc[0]);
      acc[1] = wmma4(a, b1, acc[1]);
      acc[2] = wmma4(a, b2, acc[2]);
      acc[3] = wmma4(a, b3, acc[3]);
    }
#pragma unroll
    for (int nt = 0; nt < 4; ++nt) {
      int nn = nt * 16 + l16;
#pragma unroll
      for (int i = 0; i < 8; ++i) {
        int mm = wave * 16 + i + 8 * hi;
        YT[nn * 132 + mm] = acc[nt][i] + BO[mm];
      }
    }
  }
  __syncthreads();

  // ---- LayerNorm stats over 128 channels per pixel (two-pass, biased var, eps 1e-5)
  if (tid < 64) {
    const float* row = &YT[tid * 132];
    float s = 0.f;
    for (int c = 0; c < 128; ++c) s += row[c];
    float mean = s * (1.f / 128.f);
    float v = 0.f;
    for (int c = 0; c < 128; ++c) { float d = row[c] - mean; v += d * d; }
    MEANv[tid] = mean;
    RSTDv[tid] = rsqrtf(v * (1.f / 128.f) + 1e-5f);
  }
  __syncthreads();

  // ---- Normalize, apply g, write coalesced (8 contiguous pixels per thread-iter)
  for (int it = 0; it < 4; ++it) {
    int p = tid + 256 * it;
    int c = p >> 3, r = p & 7;
    float gs = GS[c];
    float vals[8];
#pragma unroll
    for (int col = 0; col < 8; ++col) {
      int n = r * 8 + col;
      vals[col] = (YT[n * 132 + c] - MEANv[n]) * RSTDv[n] * gs;
    }
    float4* dst = (float4*)(out + gbase + (long)c * 4096 + (long)r * 64);
    dst[0] = make_float4(vals[0], vals[1], vals[2], vals[3]);
    dst[1] = make_float4(vals[4], vals[5], vals[6], vals[7]);
  }
}

extern "C" void kernel_launch(void* const* d_in, const int* in_sizes, int n_in,
                              void* d_out, int out_size, void* d_ws, size_t ws_size,
                              hipStream_t stream) {
  const float* x    = (const float*)d_in[0];
  const float* Wqkv = (const float*)d_in[1];
  const float* Wout = (const float*)d_in[2];
  const float* bout = (const float*)d_in[3];
  const float* g    = (const float*)d_in[4];
  float* out = (float*)d_out;
  (void)in_sizes; (void)n_in; (void)out_size; (void)d_ws; (void)ws_size;

  // >64KB dynamic LDS needs the attribute bump (deterministic, capture-safe).
  hipFuncSetAttribute((const void*)fused_linattn_kernel,
                      hipFuncAttributeMaxDynamicSharedMemorySize, SMEM_BYTES);

  // grid: (64 patches of 8x8, 64 batches); 256 threads = 8 wave32 per WG
  fused_linattn_kernel<<<dim3(64, 64), 256, SMEM_BYTES, stream>>>(x, Wqkv, Wout, bout, g, out);
}